// AttnModel_34273839022102
// MI455X (gfx1250) — compile-verified
//
#include <hip/hip_runtime.h>

typedef float v2f __attribute__((ext_vector_type(2)));
typedef float v8f __attribute__((ext_vector_type(8)));

#define UU 1024
#define DD 32

// One block per (batch, pair). Computes the row-0 slice of bi_attn(f1,f2):
//   bi[bp][0:32]  = O1[0,:] * f1[0,:]
//   bi[bp][32:64] = O2[0,:] * f2[0,:]
__global__ __launch_bounds__(256)
void bi_attn_row0_kernel(const float* __restrict__ a,
                         const float* __restrict__ v,
                         const float* __restrict__ l,
                         float* __restrict__ bi)   // [96][64]
{
    const int bp = blockIdx.x;          // 0..95
    const int b = bp / 3, pair = bp % 3;
    const float* f1; const float* f2;
    if (pair == 0)      { f1 = a; f2 = v; }
    else if (pair == 1) { f1 = a; f2 = l; }
    else                { f1 = v; f2 = l; }
    f1 += (size_t)b * UU * DD;
    f2 += (size_t)b * UU * DD;

    __shared__ float s_col[UU];   // sum_i exp(S[i,j])
    __shared__ float s_row[UU];   // sum_j exp(S[i,j])
    __shared__ float s_t0[UU];    // S[0,j] -> later w1[j]
    __shared__ float s_c0[UU];    // S[i,0] -> later w2[i]

    const int tid  = threadIdx.x;
    for (int i = tid; i < UU; i += 256) { s_col[i] = 0.f; s_row[i] = 0.f; }
    __syncthreads();

    const int wave  = tid >> 5;
    const int lane  = tid & 31;
    const int lrow  = lane & 15;    // row within 16-lane half (M / N index)
    const int hh    = lane >> 4;    // half select
    const int kbase = hh * 2;       // K pair handled by this half (f32 16x4 A layout)

    for (int tb = 0; tb < 8; ++tb) {
        const int ti = wave + 8 * tb;        // row block owned by this wave

        // A tile 16x32 held in registers: lane holds row ti*16+lrow,
        // K = kbase + {0,1} + 4*k  (ISA 7.12.2 32-bit A 16x4 layout, chained over 8 steps)
        v2f A[8];
        const float* arow = f1 + (size_t)(ti * 16 + lrow) * DD + kbase;
        #pragma unroll
        for (int k = 0; k < 8; ++k) A[k] = *(const v2f*)(arow + 4 * k);

        float rowAcc[8];
        #pragma unroll
        for (int vv = 0; vv < 8; ++vv) rowAcc[vv] = 0.f;

        for (int tj = 0; tj < 64; ++tj) {
            // B tile = f2 rows tj*16.. (S = f1 * f2^T), same per-lane pattern as A
            v2f Bm[8];
            const float* brow = f2 + (size_t)(tj * 16 + lrow) * DD + kbase;
            #pragma unroll
            for (int k = 0; k < 8; ++k) Bm[k] = *(const v2f*)(brow + 4 * k);

            v8f c = {0.f,0.f,0.f,0.f,0.f,0.f,0.f,0.f};
            #pragma unroll
            for (int k = 0; k < 8; ++k)
                c = __builtin_amdgcn_wmma_f32_16x16x4_f32(
                        false, A[k], false, Bm[k], (short)0, c, false, false);

            // c[vv] = S[ti*16 + vv + 8*hh, tj*16 + lrow]
            float colPartial = 0.f;
            #pragma unroll
            for (int vv = 0; vv < 8; ++vv) {
                const float e = __expf(c[vv]);
                rowAcc[vv] += e;
                colPartial += e;
            }
            atomicAdd(&s_col[tj * 16 + lrow], colPartial);   // ds_add_f32

            if (ti == 0 && hh == 0) s_t0[tj * 16 + lrow] = c[0];      // S[0, j]
            if (tj == 0 && lrow == 0) {
                #pragma unroll
                for (int vv = 0; vv < 8; ++vv)
                    s_c0[ti * 16 + vv + 8 * hh] = c[vv];              // S[i, 0]
            }
        }

        // Reduce row accumulators across the 16 lanes of each half (columns vary per lane)
        #pragma unroll
        for (int vv = 0; vv < 8; ++vv) {
            float r = rowAcc[vv];
            r += __shfl_xor(r, 1, 32);
            r += __shfl_xor(r, 2, 32);
            r += __shfl_xor(r, 4, 32);
            r += __shfl_xor(r, 8, 32);
            if (lrow == 0) s_row[ti * 16 + vv + 8 * hh] = r;
        }
    }
    __syncthreads();

    // Softmax weights for row-0 outputs (max-free: mathematically identical)
    for (int j = tid; j < UU; j += 256) {
        s_t0[j] = __expf(s_t0[j]) / s_col[j];   // w1[j]
        s_c0[j] = __expf(s_c0[j]) / s_row[j];   // w2[i]
    }
    __syncthreads();

    // O1[d] = sum_j w1[j] f2[j,d];  O2[d] = sum_i w2[i] f1[i,d]
    const int d = tid & 31;
    const int g = tid >> 5;          // 8 groups of 128 rows each
    float p1 = 0.f, p2 = 0.f;
    for (int j = g * 128; j < (g + 1) * 128; ++j) {
        p1 += s_t0[j] * f2[(size_t)j * DD + d];
        p2 += s_c0[j] * f1[(size_t)j * DD + d];
    }
    s_col[g * 32 + d] = p1;          // reuse as reduction scratch (readers done above)
    s_row[g * 32 + d] = p2;
    __syncthreads();

    if (tid < 32) {
        float o1 = 0.f, o2 = 0.f;
        #pragma unroll
        for (int gg = 0; gg < 8; ++gg) { o1 += s_col[gg * 32 + tid]; o2 += s_row[gg * 32 + tid]; }
        float* outp = bi + bp * 64;
        outp[tid]      = o1 * f1[tid];   // O1 * f1[b,0,:]
        outp[32 + tid] = o2 * f2[tid];   // O2 * f2[b,0,:]
    }
}

// fc1 + tanh + fc2, softmax over batch, alpha-mix -> per-batch 64-wide row
__global__ __launch_bounds__(128)
void head_kernel(const float* __restrict__ bi,     // [96][64]  (b*3+k)
                 const float* __restrict__ fc1_w,  // [64][64]
                 const float* __restrict__ fc1_b,  // [64]
                 const float* __restrict__ fc2_w,  // [64]
                 float* __restrict__ row)          // [32][64]
{
    __shared__ float s_ci[96];
    __shared__ float s_alpha[96];
    const int t = threadIdx.x;

    if (t < 96) {
        const float* x = bi + t * 64;
        float ci = 0.f;
        for (int o = 0; o < 64; ++o) {
            float acc = fc1_b[o];
            #pragma unroll 4
            for (int dd = 0; dd < 64; ++dd) acc += x[dd] * fc1_w[o * 64 + dd];
            ci += tanhf(acc) * fc2_w[o];
        }
        s_ci[t] = ci;
    }
    __syncthreads();
    if (t < 96) {
        const int k = t % 3;                 // softmax over batch dim per k
        float m = -1e30f;
        for (int bb = 0; bb < 32; ++bb) m = fmaxf(m, s_ci[bb * 3 + k]);
        float s = 0.f;
        for (int bb = 0; bb < 32; ++bb) s += __expf(s_ci[bb * 3 + k] - m);
        s_alpha[t] = __expf(s_ci[t] - m) / s;
    }
    __syncthreads();
    for (int idx = t; idx < 32 * 64; idx += 128) {
        const int bb = idx >> 6, dd = idx & 63;
        float acc = 0.f;
        #pragma unroll
        for (int k = 0; k < 3; ++k)
            acc += s_alpha[bb * 3 + k] * bi[(bb * 3 + k) * 64 + dd];
        row[idx] = acc;
    }
}

// Tile each batch's 64-float row over u=1024 -> [32][1024][64] output (8 MB stream)
__global__ __launch_bounds__(256)
void bcast_kernel(const float* __restrict__ row, float* __restrict__ out)
{
    const int idx = blockIdx.x * 256 + threadIdx.x;   // float4 index; 524288 total
    const int b  = idx >> 14;                         // 16384 float4 per batch
    const int d4 = idx & 15;                          // 16 float4 per 64-float row
    const float4 val = ((const float4*)row)[b * 16 + d4];
    ((float4*)out)[idx] = val;
}

extern "C" void kernel_launch(void* const* d_in, const int* in_sizes, int n_in,
                              void* d_out, int out_size, void* d_ws, size_t ws_size,
                              hipStream_t stream) {
    const float* a_emb = (const float*)d_in[0];
    const float* v_emb = (const float*)d_in[1];
    const float* l_emb = (const float*)d_in[2];
    const float* fc1_w = (const float*)d_in[3];
    const float* fc1_b = (const float*)d_in[4];
    const float* fc2_w = (const float*)d_in[5];
    float* out = (float*)d_out;

    float* bi  = (float*)d_ws;        // 96*64 floats
    float* row = bi + 96 * 64;        // 32*64 floats

    bi_attn_row0_kernel<<<96, 256, 0, stream>>>(a_emb, v_emb, l_emb, bi);
    head_kernel<<<1, 128, 0, stream>>>(bi, fc1_w, fc1_b, fc2_w, row);
    bcast_kernel<<<(32 * 1024 * 64 / 4) / 256, 256, 0, stream>>>(row, out);
}